// Cyto3ONNX_45208825757987
// MI455X (gfx1250) — compile-verified
//
#include <hip/hip_runtime.h>

// Cellpose-style flow following on MI455X (gfx1250).
//
//   prep_border : zero the 1-pixel ring of the padded field.
//   prep_rows   : row-per-block build of the zero-padded, channel-interleaved
//                 field im2[(y+1)*1026 + (x+1)] = {dP0*m, dP1*m},
//                 m = (cellprob>0)*0.2.  Uses gfx1250 async global->LDS b128
//                 DMA (ASYNCcnt + s_wait_asynccnt) when the toolchain exposes
//                 the builtin; direct b128 register loads otherwise.
//   chase_pairs : each thread advects TWO points (two independent dependency
//                 chains per wave32 lane). Per step: 4x global_load_b64 from
//                 one base (+imm offsets 0/8/8208/8216) out of L2 (field is
//                 8.4 MB << 192 MB L2), 12 FMAs, clamp. All iteration math in
//                 padded-pixel space; one affine back-transform at the end.

namespace {

constexpr int kH  = 1024;
constexpr int kW  = 1024;
constexpr int kPW = kW + 2;          // padded width
constexpr int kPH = kH + 2;          // padded height
constexpr int kHW = kH * kW;

typedef float v4f __attribute__((ext_vector_type(4)));
typedef int   v4i __attribute__((ext_vector_type(4)));

#if defined(__gfx1250__) && __has_builtin(__builtin_amdgcn_global_load_async_to_lds_b128)
#define PREP_ASYNC_LDS 1
#endif

// ------------------------------------------------------------- border ----
__global__ __launch_bounds__(256) void prep_border(float2* __restrict__ im2)
{
    int t = blockIdx.x * 256 + threadIdx.x;
    const float2 z = make_float2(0.0f, 0.0f);
    if (t < kPW) {
        im2[t] = z;                                        // top row
    } else if (t < 2 * kPW) {
        im2[(kPH - 1) * kPW + (t - kPW)] = z;              // bottom row
    } else if (t < 2 * kPW + kH) {
        im2[(t - 2 * kPW + 1) * kPW] = z;                  // left column
    } else if (t < 2 * kPW + 2 * kH) {
        im2[(t - 2 * kPW - kH + 1) * kPW + (kPW - 1)] = z; // right column
    }
}

// --------------------------------------------------------------- rows ----
// grid.x = kH (one interior row per block), 256 threads, 4 pixels/thread.
__global__ __launch_bounds__(256) void prep_rows(
    const float* __restrict__ dP,        // [2, H, W]
    const float* __restrict__ cellprob,  // [H, W]
    float2* __restrict__ im2)            // [PH, PW]
{
    const int r = blockIdx.x;            // interior row 0..kH-1
    const int i = threadIdx.x;           // 4 consecutive pixels per thread

    float d0[4], d1[4], cp[4];

#if PREP_ASYNC_LDS
    // gfx1250 async DMA: three b128 lane-transfers into LDS (ASYNCcnt),
    // one s_wait_asynccnt, no barrier (each lane consumes only its own 16B).
    __shared__ float lds[3 * kW];
    // Builtin signature (from clang diagnostic):
    //   (v4i addrspace(1)* src, v4i addrspace(3)* dst, Imm offset, Imm cpol)
    typedef __attribute__((address_space(1))) v4i* gv4i_p;
    typedef __attribute__((address_space(3))) v4i* lv4i_p;

    const float* g0 = dP + (size_t)r * kW + 4 * i;
    const float* g1 = dP + (size_t)kHW + (size_t)r * kW + 4 * i;
    const float* gc = cellprob + (size_t)r * kW + 4 * i;
    gv4i_p gp0 = (gv4i_p)(uintptr_t)g0;
    gv4i_p gp1 = (gv4i_p)(uintptr_t)g1;
    gv4i_p gpc = (gv4i_p)(uintptr_t)gc;
    // LDS byte offset = low 32 bits of the generic (aperture-based) address.
    lv4i_p l0 = (lv4i_p)(unsigned)(uintptr_t)(lds + 4 * i);
    lv4i_p l1 = (lv4i_p)(unsigned)(uintptr_t)(lds + kW + 4 * i);
    lv4i_p lc = (lv4i_p)(unsigned)(uintptr_t)(lds + 2 * kW + 4 * i);

    __builtin_amdgcn_global_load_async_to_lds_b128(gp0, l0, 0, 0);
    __builtin_amdgcn_global_load_async_to_lds_b128(gp1, l1, 0, 0);
    __builtin_amdgcn_global_load_async_to_lds_b128(gpc, lc, 0, 0);
#if __has_builtin(__builtin_amdgcn_s_wait_asynccnt)
    __builtin_amdgcn_s_wait_asynccnt(0);
#else
    asm volatile("s_wait_asynccnt 0x0" ::: "memory");
#endif
    for (int k = 0; k < 4; ++k) {
        d0[k] = lds[4 * i + k];
        d1[k] = lds[kW + 4 * i + k];
        cp[k] = lds[2 * kW + 4 * i + k];
    }
#else
    // Direct path: three coalesced b128 register loads.
    v4f a = ((const v4f*)(dP + (size_t)r * kW))[i];
    v4f b = ((const v4f*)(dP + (size_t)kHW + (size_t)r * kW))[i];
    v4f c = ((const v4f*)(cellprob + (size_t)r * kW))[i];
    d0[0] = a.x; d0[1] = a.y; d0[2] = a.z; d0[3] = a.w;
    d1[0] = b.x; d1[1] = b.y; d1[2] = b.z; d1[3] = b.w;
    cp[0] = c.x; cp[1] = c.y; cp[2] = c.z; cp[3] = c.w;
#endif

    float2* __restrict__ orow = im2 + (size_t)(r + 1) * kPW + 1 + 4 * i;
#pragma unroll
    for (int k = 0; k < 4; ++k) {
        float m = (cp[k] > 0.0f) ? 0.2f : 0.0f;
        orow[k] = make_float2(d0[k] * m, d1[k] * m);
    }
}

// --------------------------------------------------------------- chase ----
// Padded pixel coords: p = ((n + 1)*S - 1)/2 + 1 = n*S/2 + S/2 + 0.5
// clip(n, -1, 1)  <=>  clamp(p, 0.5, S + 0.5);  back: n = (2p - 1)/S - 1
__global__ __launch_bounds__(256) void chase_pairs(
    const float2* __restrict__ im2,      // [PH, PW]
    const v4f*    __restrict__ pt0,      // [H*W/2]: two (x,y) points each
    v4f*          __restrict__ out,
    const int*    __restrict__ niter_p)
{
    int t = blockIdx.x * 256 + threadIdx.x;
    if (t >= kHW / 2) return;
    const int niter = niter_p[0];        // uniform scalar load

    v4f p = __builtin_nontemporal_load(&pt0[t]);   // streamed once
    float ax = fmaf(p.x, 512.0f, 512.5f);
    float ay = fmaf(p.y, 512.0f, 512.5f);
    float bx = fmaf(p.z, 512.0f, 512.5f);
    float by = fmaf(p.w, 512.0f, 512.5f);

    for (int i = 0; i < niter; ++i) {
        float fax = floorf(ax), fay = floorf(ay);
        float fbx = floorf(bx), fby = floorf(by);
        float wax = ax - fax,   way = ay - fay;
        float wbx = bx - fbx,   wby = by - fby;
        const float2* pa = im2 + ((int)fay * kPW + (int)fax);
        const float2* pb = im2 + ((int)fby * kPW + (int)fbx);

        // 8 independent b64 gathers (one base each, imm offsets)
        float2 a00 = pa[0], a01 = pa[1], a10 = pa[kPW], a11 = pa[kPW + 1];
        float2 b00 = pb[0], b01 = pb[1], b10 = pb[kPW], b11 = pb[kPW + 1];

        float atx = fmaf(wax, a01.x - a00.x, a00.x);
        float abx = fmaf(wax, a11.x - a10.x, a10.x);
        float dxa = fmaf(way, abx - atx, atx);
        float aty = fmaf(wax, a01.y - a00.y, a00.y);
        float aby = fmaf(wax, a11.y - a10.y, a10.y);
        float dya = fmaf(way, aby - aty, aty);

        float btx = fmaf(wbx, b01.x - b00.x, b00.x);
        float bbx = fmaf(wbx, b11.x - b10.x, b10.x);
        float dxb = fmaf(wby, bbx - btx, btx);
        float bty = fmaf(wbx, b01.y - b00.y, b00.y);
        float bby = fmaf(wbx, b11.y - b10.y, b10.y);
        float dyb = fmaf(wby, bby - bty, bty);

        ax = fminf(fmaxf(fmaf(dxa, 512.0f, ax), 0.5f), (float)kW + 0.5f);
        ay = fminf(fmaxf(fmaf(dya, 512.0f, ay), 0.5f), (float)kH + 0.5f);
        bx = fminf(fmaxf(fmaf(dxb, 512.0f, bx), 0.5f), (float)kW + 0.5f);
        by = fminf(fmaxf(fmaf(dyb, 512.0f, by), 0.5f), (float)kH + 0.5f);
    }

    v4f r;
    r.x = fmaf(ax, 0.001953125f, -1.0009765625f);
    r.y = fmaf(ay, 0.001953125f, -1.0009765625f);
    r.z = fmaf(bx, 0.001953125f, -1.0009765625f);
    r.w = fmaf(by, 0.001953125f, -1.0009765625f);
    __builtin_nontemporal_store(r, &out[t]);       // streamed once
}

// ----------------------------------------------- fallback (no workspace) --
__device__ __forceinline__ float2 sample_raw(
    const float* __restrict__ dP, const float* __restrict__ cp, int x, int y)
{
    float2 v = make_float2(0.0f, 0.0f);
    if ((unsigned)x < (unsigned)kW && (unsigned)y < (unsigned)kH) {
        int idx = y * kW + x;
        float m = (cp[idx] > 0.0f) ? 0.2f : 0.0f;
        v.x = dP[idx] * m;
        v.y = dP[kHW + idx] * m;
    }
    return v;
}

__global__ __launch_bounds__(256) void chase_fallback(
    const float* __restrict__ dP, const float* __restrict__ cp,
    const float2* __restrict__ pt0, float2* __restrict__ out,
    const int* __restrict__ niter_p)
{
    int t = blockIdx.x * 256 + threadIdx.x;
    if (t >= kHW) return;
    const int niter = niter_p[0];
    float2 p = pt0[t];
    float x = fmaf(p.x, 512.0f, 511.5f);
    float y = fmaf(p.y, 512.0f, 511.5f);
    for (int i = 0; i < niter; ++i) {
        float fx = floorf(x), fy = floorf(y);
        float wx = x - fx,    wy = y - fy;
        int ix = (int)fx, iy = (int)fy;
        float2 v00 = sample_raw(dP, cp, ix,     iy);
        float2 v01 = sample_raw(dP, cp, ix + 1, iy);
        float2 v10 = sample_raw(dP, cp, ix,     iy + 1);
        float2 v11 = sample_raw(dP, cp, ix + 1, iy + 1);
        float tx = fmaf(wx, v01.x - v00.x, v00.x);
        float bxv = fmaf(wx, v11.x - v10.x, v10.x);
        float dx = fmaf(wy, bxv - tx, tx);
        float ty = fmaf(wx, v01.y - v00.y, v00.y);
        float byv = fmaf(wx, v11.y - v10.y, v10.y);
        float dy = fmaf(wy, byv - ty, ty);
        x = fminf(fmaxf(fmaf(dx, 512.0f, x), -0.5f), 1023.5f);
        y = fminf(fmaxf(fmaf(dy, 512.0f, y), -0.5f), 1023.5f);
    }
    out[t] = make_float2(fmaf(x, 0.001953125f, -0.9990234375f),
                         fmaf(y, 0.001953125f, -0.9990234375f));
}

} // namespace

extern "C" void kernel_launch(void* const* d_in, const int* in_sizes, int n_in,
                              void* d_out, int out_size, void* d_ws, size_t ws_size,
                              hipStream_t stream) {
    const float* dP       = (const float*)d_in[0];  // [2,1024,1024]
    const float* cellprob = (const float*)d_in[1];  // [1024,1024]
    const float* pt0      = (const float*)d_in[2];  // [1024,1024,2]
    const int*   niter    = (const int*)d_in[3];    // scalar (device)

    const size_t need = (size_t)kPW * kPH * sizeof(float2);
    if (ws_size >= need) {
        float2* im2 = (float2*)d_ws;
        {
            int n = 2 * kPW + 2 * kH;   // padded ring
            prep_border<<<(n + 255) / 256, 256, 0, stream>>>(im2);
        }
        prep_rows<<<kH, 256, 0, stream>>>(dP, cellprob, im2);
        {
            int n = kHW / 2;            // two points per thread
            chase_pairs<<<(n + 255) / 256, 256, 0, stream>>>(
                im2, (const v4f*)pt0, (v4f*)d_out, niter);
        }
    } else {
        int n = kHW;
        chase_fallback<<<(n + 255) / 256, 256, 0, stream>>>(
            dP, cellprob, (const float2*)pt0, (float2*)d_out, niter);
    }
}